// BasicEuclideanDistModel_6373731467457
// MI455X (gfx1250) — compile-verified
//
#include <hip/hip_runtime.h>
#include <math.h>

#define NRIEMANN 128
#define EPS_F 1e-6f
#define EV_BLOCKS 2048
#define BLOCK 256

typedef __attribute__((ext_vector_type(2))) float v2f;
typedef __attribute__((ext_vector_type(8))) float v8f;

// Wave32 reduction via V_WMMA_F32_16X16X4_F32:
// A (16x4 f32, 2 VGPRs/lane) holds 2 accumulators per lane, B = ones.
// D[m][n] = rowsum(A_m); per-lane sum of 8 D VGPRs = rows 0-7 (lanes 0-15)
// or rows 8-15 (lanes 16-31); xor-16 shuffle completes the 64-value sum.
__device__ __forceinline__ float wave_sum_wmma(float a0, float a1) {
    v2f A; A[0] = a0;   A[1] = a1;
    v2f B; B[0] = 1.0f; B[1] = 1.0f;
    v8f C = {};
    v8f D = __builtin_amdgcn_wmma_f32_16x16x4_f32(
        /*neg_a=*/false, A, /*neg_b=*/false, B,
        /*c_mod=*/(short)0, C, /*reuse_a=*/false, /*reuse_b=*/false);
    float s = ((D[0] + D[1]) + (D[2] + D[3])) + ((D[4] + D[5]) + (D[6] + D[7]));
    s += __shfl_xor(s, 16, 32);
    return s;  // every lane holds the wave total
}

__device__ __forceinline__ float block_sum(float a0, float a1) {
    __shared__ float lds[BLOCK / 32];
    float s = wave_sum_wmma(a0, a1);
    int lane = threadIdx.x & 31;
    int wave = threadIdx.x >> 5;
    if (lane == 0) lds[wave] = s;
    __syncthreads();
    float r = 0.0f;
    if (threadIdx.x == 0) {
        r = ((lds[0] + lds[1]) + (lds[2] + lds[3])) +
            ((lds[4] + lds[5]) + (lds[6] + lds[7]));
    }
    return r;
}

// ---- Event term: sum over events of |z_u(t) - z_v(t) + eps| ----
__global__ void __launch_bounds__(BLOCK)
ev_kernel(const float* __restrict__ z0, const float* __restrict__ v0,
          const float* __restrict__ a0, const int* __restrict__ u,
          const int* __restrict__ v, const float* __restrict__ et,
          int n, float* __restrict__ part) {
    const float2* z2 = (const float2*)z0;
    const float2* v2 = (const float2*)v0;
    const float2* a2 = (const float2*)a0;
    float acc = 0.0f;
    int stride = gridDim.x * blockDim.x;
    for (int e = blockIdx.x * blockDim.x + threadIdx.x; e < n; e += stride) {
        int ui = u[e];
        int vi = v[e];
        float t = et[e];
        float h = 0.5f * t * t;
        float2 zu = z2[ui], vu = v2[ui], au = a2[ui];
        float2 zv = z2[vi], vv = v2[vi], av = a2[vi];
        float dx = (zu.x + vu.x * t + au.x * h) - (zv.x + vv.x * t + av.x * h) + EPS_F;
        float dy = (zu.y + vu.y * t + au.y * h) - (zv.y + vv.y * t + av.y * h) + EPS_F;
        acc += sqrtf(dx * dx + dy * dy);
    }
    float bs = block_sum(acc, 0.0f);
    if (threadIdx.x == 0) part[blockIdx.x] = bs;
}

// ---- Non-event term: sum over pairs x 128 steps of exp(b - |diff(t)|) ----
__global__ void __launch_bounds__(BLOCK)
ne_kernel(const float* __restrict__ beta, const float* __restrict__ z0,
          const float* __restrict__ v0, const float* __restrict__ a0,
          const int* __restrict__ pu, const int* __restrict__ pv,
          const float* __restrict__ t0p, const float* __restrict__ tnp,
          int n, float* __restrict__ part) {
    const float b  = beta[0];
    const float t0 = t0p[0];
    const float tn = tnp[0];
    const float step = (tn - t0) * (1.0f / (float)NRIEMANN);

    float acc0 = 0.0f, acc1 = 0.0f;
    int p = blockIdx.x * blockDim.x + threadIdx.x;
    if (p < n) {
        const float2* z2 = (const float2*)z0;
        const float2* v2 = (const float2*)v0;
        const float2* a2 = (const float2*)a0;
        int ui = pu[p];
        int vi = pv[p];
        float2 zu = z2[ui], zv = z2[vi];
        float2 vu = v2[ui], vv = v2[vi];
        float2 au = a2[ui], av = a2[vi];
        float dzx = zu.x - zv.x, dzy = zu.y - zv.y;
        float dvx = vu.x - vv.x, dvy = vu.y - vv.y;
        float dax = 0.5f * (au.x - av.x), day = 0.5f * (au.y - av.y);
#pragma unroll 4
        for (int i = 0; i < NRIEMANN / 2; ++i) {
            float t  = t0 + (float)i * step;
            float fx = dzx + t * dvx + t * t * dax + EPS_F;
            float fy = dzy + t * dvy + t * t * day + EPS_F;
            acc0 += __expf(b - sqrtf(fx * fx + fy * fy));

            float s  = t0 + (float)(i + NRIEMANN / 2) * step;
            float gx = dzx + s * dvx + s * s * dax + EPS_F;
            float gy = dzy + s * dvy + s * s * day + EPS_F;
            acc1 += __expf(b - sqrtf(gx * gx + gy * gy));
        }
    }
    float bs = block_sum(acc0, acc1);
    if (threadIdx.x == 0) part[blockIdx.x] = bs;
}

// ---- Final combine: f64 deterministic reduction of block partials ----
__global__ void __launch_bounds__(BLOCK)
fin_kernel(const float* __restrict__ part_ev, int n_ev_part,
           const float* __restrict__ part_ne, int n_ne_part,
           const float* __restrict__ beta, const float* __restrict__ t0p,
           const float* __restrict__ tnp, int n_events,
           float* __restrict__ out) {
    __shared__ double l1[BLOCK];
    __shared__ double l2[BLOCK];
    double s1 = 0.0, s2 = 0.0;
    for (int i = threadIdx.x; i < n_ev_part; i += BLOCK) s1 += (double)part_ev[i];
    for (int i = threadIdx.x; i < n_ne_part; i += BLOCK) s2 += (double)part_ne[i];
    l1[threadIdx.x] = s1;
    l2[threadIdx.x] = s2;
    __syncthreads();
    for (int off = BLOCK / 2; off > 0; off >>= 1) {
        if ((int)threadIdx.x < off) {
            l1[threadIdx.x] += l1[threadIdx.x + off];
            l2[threadIdx.x] += l2[threadIdx.x + off];
        }
        __syncthreads();
    }
    if (threadIdx.x == 0) {
        double b  = (double)beta[0];
        double dt = ((double)tnp[0] - (double)t0p[0]) / (double)NRIEMANN;
        double event_intensity = b * (double)n_events - l1[0];
        double non_event_intensity = l2[0] * dt;
        out[0] = (float)(event_intensity - non_event_intensity);
    }
}

extern "C" void kernel_launch(void* const* d_in, const int* in_sizes, int n_in,
                              void* d_out, int out_size, void* d_ws, size_t ws_size,
                              hipStream_t stream) {
    const float* beta = (const float*)d_in[0];
    const float* z0   = (const float*)d_in[1];
    const float* v0   = (const float*)d_in[2];
    const float* a0   = (const float*)d_in[3];
    const int*   u    = (const int*)d_in[4];
    const int*   v    = (const int*)d_in[5];
    const float* et   = (const float*)d_in[6];
    const int*   pu   = (const int*)d_in[7];
    const int*   pv   = (const int*)d_in[8];
    const float* t0p  = (const float*)d_in[9];
    const float* tnp  = (const float*)d_in[10];

    int n_events = in_sizes[6];
    int n_pairs  = in_sizes[7];

    int nb_ev = EV_BLOCKS;
    int nb_ne = (n_pairs + BLOCK - 1) / BLOCK;

    float* part_ev = (float*)d_ws;
    float* part_ne = part_ev + nb_ev;

    ev_kernel<<<nb_ev, BLOCK, 0, stream>>>(z0, v0, a0, u, v, et, n_events, part_ev);
    ne_kernel<<<nb_ne, BLOCK, 0, stream>>>(beta, z0, v0, a0, pu, pv, t0p, tnp,
                                           n_pairs, part_ne);
    fin_kernel<<<1, BLOCK, 0, stream>>>(part_ev, nb_ev, part_ne, nb_ne,
                                        beta, t0p, tnp, n_events, (float*)d_out);
}